// CRFDependency_5514738008072
// MI455X (gfx1250) — compile-verified
//
#include <hip/hip_runtime.h>
#include <hip/hip_bf16.h>
#include <stdint.h>

#define NEGV   (-1e30f)
#define NN     128
#define BB     128
#define NTHREADS 1024
#define NWAVES   32   // wave32: 1024 threads = 32 waves

typedef __attribute__((ext_vector_type(4))) unsigned int u32x4;
typedef __attribute__((ext_vector_type(8))) unsigned int u32x8;
typedef __attribute__((ext_vector_type(2))) float v2f;
typedef __attribute__((ext_vector_type(8))) float v8f;

__device__ __forceinline__ float wave_max(float m) {
  #pragma unroll
  for (int off = 16; off; off >>= 1)
    m = fmaxf(m, __shfl_xor(m, off, 32));
  return m;
}
__device__ __forceinline__ float wave_sum(float s) {
  #pragma unroll
  for (int off = 16; off; off >>= 1)
    s += __shfl_xor(s, off, 32);
  return s;
}

// One workgroup per batch. Entire DP runs in LDS:
//   sC  [16384] : s_c[i][j]
//   sCT [16384] : s_c transposed (kills 32-way bank conflicts on column reads)
//   sI  [16384] : s_i[i][j]
//   sS  [16384] : scores[b][i][j], DMA'd in by the Tensor Data Mover
// Total 256KB <= 320KB WGP LDS.
__global__ __launch_bounds__(NTHREADS)
void crf_inside_kernel(const float* __restrict__ scores,
                       const unsigned char* __restrict__ mask,
                       float* __restrict__ partial) {
  extern __shared__ float smem[];
  float* sC  = smem;
  float* sCT = smem + 16384;
  float* sI  = smem + 2 * 16384;
  float* sS  = smem + 3 * 16384;
  __shared__ int sLen;

  const int b    = blockIdx.x;
  const int tid  = threadIdx.x;
  const int wid  = tid >> 5;
  const int lane = tid & 31;

  // ---- init DP state: s_c = NEG except diag 0, s_i = NEG ----
  for (int i = tid; i < 16384; i += NTHREADS) {
    float v = ((i >> 7) == (i & 127)) ? 0.0f : NEGV;
    sC[i]  = v;
    sCT[i] = v;
    sI[i]  = NEGV;
  }

  if (tid == 0) {
    sLen = 0;
    // ---- TDM: DMA scores[b] (128x128 f32 = 64KB) global -> LDS ----
    unsigned long long ga = (unsigned long long)(const void*)(scores + (size_t)b * (NN * NN));
    unsigned lds_off = (unsigned)(uintptr_t)(void*)sS;  // low 32 bits of shared-aperture addr = LDS offset

    u32x4 g0;
    g0.x = 1u;                                            // count=1 (valid descriptor), user mode
    g0.y = lds_off;                                       // lds_addr
    g0.z = (unsigned)ga;                                  // global_addr[31:0]
    g0.w = (unsigned)((ga >> 32) & 0x01FFFFFFull)         // global_addr[56:32]
         | 0x80000000u;                                   // type=2 ("image")

    u32x8 g1;
    g1[0] = 0x00020000u;       // workgroup_mask=0 (not in cluster), data_size=2 -> 4 bytes
    g1[1] = (128u << 16);      // tensor_dim0 = 128 (low 16 bits at [63:48])
    g1[2] = (128u << 16);      // tensor_dim0 hi=0 ; tensor_dim1 = 128 (low 16 at [95:80])
    g1[3] = (128u << 16);      // tensor_dim1 hi=0 ; tile_dim0 = 128
    g1[4] = 128u;              // tile_dim1 = 128 ; tile_dim2 = 0
    g1[5] = 128u;              // tensor_dim0_stride = 128 elements (rows contiguous)
    g1[6] = 0u;                // dim0_stride hi, tensor_dim1_stride lo (unused, 2D tile)
    g1[7] = 0u;

    asm volatile("tensor_load_to_lds %0, %1" :: "s"(g0), "s"(g1) : "memory");
  }
  __syncthreads();  // sLen=0 visible; TDM writes target sS only (disjoint from init)

  // lens[b] = popcount of mask row (mask is jnp.bool_, 1 byte/elem), 128 parallel loads
  if (tid < NN) {
    if (mask[(size_t)b * NN + tid]) atomicAdd(&sLen, 1);
  }
  if (tid == 0) __builtin_amdgcn_s_wait_tensorcnt(0);
  __syncthreads();  // sS and sLen ready for all waves

  // ---- inside recursion over span widths ----
  for (int w = 1; w < NN; ++w) {
    const int n = NN - w;
    for (int k = wid; k < n; k += NWAVES) {
      const int rowK  = k << 7;
      const int rowKW = (k + w) << 7;

      // red1: il = lse_l( s_c[k, k+l] + s_c[k+w, k+1+l] ),  l in [0,w)
      float x1[4];
      float m = -3.0e38f;
      #pragma unroll
      for (int c = 0; c < 4; ++c) {
        const int l = lane + (c << 5);
        if (l < w) {
          x1[c] = sC[rowK + k + l] + sC[rowKW + k + 1 + l];
          m = fmaxf(m, x1[c]);
        }
      }
      m = wave_max(m);
      float s = 0.0f;
      #pragma unroll
      for (int c = 0; c < 4; ++c) {
        const int l = lane + (c << 5);
        if (l < w) s += __expf(x1[c] - m);
      }
      s = wave_sum(s);
      const float il = m + __logf(s);

      // same-iteration s_i values kept in registers (avoid LDS RAW inside step)
      const float siL = il + sS[rowK + k + w];   // s_i[k+w, k] = il + scores[b,k,k+w]
      const float siR = il + sS[rowKW + k];      // s_i[k, k+w] = il + scores[b,k+w,k]

      // red2: cl = lse_l( s_c[k+l, k] + s_i[k+w, k+l] ); l=0 term = 0 + siL
      float x2[4];
      float m2 = -3.0e38f;
      #pragma unroll
      for (int c = 0; c < 4; ++c) {
        const int l = lane + (c << 5);
        if (l < w) {
          x2[c] = (l == 0) ? siL : (sCT[rowK + k + l] + sI[rowKW + k + l]);
          m2 = fmaxf(m2, x2[c]);
        }
      }
      m2 = wave_max(m2);
      float s2 = 0.0f;
      #pragma unroll
      for (int c = 0; c < 4; ++c) {
        const int l = lane + (c << 5);
        if (l < w) s2 += __expf(x2[c] - m2);
      }
      s2 = wave_sum(s2);
      const float clv = m2 + __logf(s2);

      // red3: cr = lse_l( s_i[k, k+1+l] + s_c[k+1+l, k+w] ); l=w-1 term = siR + 0
      float x3[4];
      float m3 = -3.0e38f;
      #pragma unroll
      for (int c = 0; c < 4; ++c) {
        const int l = lane + (c << 5);
        if (l < w) {
          x3[c] = (l == w - 1) ? siR : (sI[rowK + k + 1 + l] + sCT[rowKW + k + 1 + l]);
          m3 = fmaxf(m3, x3[c]);
        }
      }
      m3 = wave_max(m3);
      float s3 = 0.0f;
      #pragma unroll
      for (int c = 0; c < 4; ++c) {
        const int l = lane + (c << 5);
        if (l < w) s3 += __expf(x3[c] - m3);
      }
      s3 = wave_sum(s3);
      float crv = m3 + __logf(s3);

      if (lane == 0) {
        sI[rowKW + k]     = siL;                        // s_i[k+w, k]
        sI[rowK + k + w]  = siR;                        // s_i[k, k+w]
        sC[rowKW + k]     = clv;                        // s_c[k+w, k]
        sCT[rowK + k + w] = clv;
        if (k == 0 && sLen != w) crv = NEGV;            // mask s_c[0, w] by (lens == w)
        sC[rowK + k + w]  = crv;                        // s_c[k, k+w]
        sCT[rowKW + k]    = crv;
      }
    }
    __syncthreads();
  }

  if (tid == 0) partial[b] = sC[sLen];                  // s_c[0, lens[b]]
}

// Deterministic sum of 128 per-batch partials on the matrix pipe.
// Two V_WMMA_F32_16X16X4_F32 with B = ones accumulate row sums of the 64-value
// A fragments into D; because B is all-ones, every column of D holds the same
// 16 row sums, and (lane j) + (lane j^16) column halves give the exact total.
// Single wave32, EXEC all-ones as WMMA requires.
__global__ __launch_bounds__(32)
void crf_reduce_kernel(const float* __restrict__ partial,
                       float* __restrict__ out) {
  const int lane = threadIdx.x;  // 32 lanes
  v2f a0, a1, bo;
  a0.x = partial[lane * 2 + 0];
  a0.y = partial[lane * 2 + 1];
  a1.x = partial[64 + lane * 2 + 0];
  a1.y = partial[64 + lane * 2 + 1];
  bo.x = 1.0f;
  bo.y = 1.0f;
  v8f c = {};
  c = __builtin_amdgcn_wmma_f32_16x16x4_f32(false, a0, false, bo, (short)0, c, false, false);
  c = __builtin_amdgcn_wmma_f32_16x16x4_f32(false, a1, false, bo, (short)0, c, false, false);
  float colsum = c[0] + c[1] + c[2] + c[3] + c[4] + c[5] + c[6] + c[7];
  float total = colsum + __shfl_xor(colsum, 16, 32);
  if (lane == 0) out[0] = total;
}

extern "C" void kernel_launch(void* const* d_in, const int* in_sizes, int n_in,
                              void* d_out, int out_size, void* d_ws, size_t ws_size,
                              hipStream_t stream) {
  (void)in_sizes; (void)n_in; (void)out_size; (void)ws_size;
  const float*         scores = (const float*)d_in[0];
  const unsigned char* mask   = (const unsigned char*)d_in[1];
  float* partial = (float*)d_ws;   // BB floats of scratch
  float* out     = (float*)d_out;

  const size_t smem = 4 * 16384 * sizeof(float);  // 256 KB dynamic LDS per workgroup
  hipFuncSetAttribute(reinterpret_cast<const void*>(crf_inside_kernel),
                      hipFuncAttributeMaxDynamicSharedMemorySize, (int)smem);

  crf_inside_kernel<<<BB, NTHREADS, smem, stream>>>(scores, mask, partial);
  crf_reduce_kernel<<<1, 32, 0, stream>>>(partial, out);
}